// IsoGCN_74019466379869
// MI455X (gfx1250) — compile-verified
//
#include <hip/hip_runtime.h>
#include <hip/hip_bf16.h>

#define FDIM 64

typedef float v2f __attribute__((ext_vector_type(2)));
typedef float v8f __attribute__((ext_vector_type(8)));

// ---------------- zero workspace (h1 + h regions) ----------------
__global__ void iso_zero(float4* __restrict__ p, long n4) {
  long i = (long)blockIdx.x * blockDim.x + threadIdx.x;
  long stride = (long)gridDim.x * blockDim.x;
  for (; i < n4; i += stride) p[i] = make_float4(0.f, 0.f, 0.f, 0.f);
}

// ---------------- pass 1: h1[r,d,f] += vals[d,e] * x[c,f] ----------------
// thread = (feature-chunk of 4, run of 16 consecutive edges). rows sorted ->
// accumulate in registers, flush on row change (atomics reduced ~16x).
__global__ void __launch_bounds__(256) iso_pass1(
    const float* __restrict__ x, const float* __restrict__ vals,
    const int* __restrict__ rows, const int* __restrict__ cols,
    float* __restrict__ h1, int nE, int nRuns) {
  int t   = blockIdx.x * blockDim.x + threadIdx.x;
  int fc  = (t & 15) * 4;          // 4-feature chunk
  int run = t >> 4;
  if (run >= nRuns) return;
  int e0 = run * 16;
  int e1 = e0 + 16; if (e1 > nE) e1 = nE;

  float a0[4] = {0,0,0,0}, a1[4] = {0,0,0,0}, a2[4] = {0,0,0,0};
  int curRow = rows[e0];
  for (int e = e0; e < e1; ++e) {
    int   r  = rows[e];
    int   c  = cols[e];
    float v0 = vals[e];
    float v1 = vals[(long)nE + e];
    float v2 = vals[2L * nE + e];
    float4 xv = *(const float4*)(x + (long)c * FDIM + fc);
    if (e + 1 < e1)  // prefetch next gathered row (random L2 access)
      __builtin_prefetch(x + (long)cols[e + 1] * FDIM + fc, 0, 0);
    if (r != curRow) {
      float* base = h1 + (long)curRow * 3 * FDIM + fc;
      #pragma unroll
      for (int j = 0; j < 4; ++j) {
        atomicAdd(base + j,            a0[j]);
        atomicAdd(base + FDIM + j,     a1[j]);
        atomicAdd(base + 2 * FDIM + j, a2[j]);
        a0[j] = a1[j] = a2[j] = 0.f;
      }
      curRow = r;
    }
    float xj[4] = {xv.x, xv.y, xv.z, xv.w};
    #pragma unroll
    for (int j = 0; j < 4; ++j) {
      a0[j] += v0 * xj[j];
      a1[j] += v1 * xj[j];
      a2[j] += v2 * xj[j];
    }
  }
  float* base = h1 + (long)curRow * 3 * FDIM + fc;
  #pragma unroll
  for (int j = 0; j < 4; ++j) {
    atomicAdd(base + j,            a0[j]);
    atomicAdd(base + FDIM + j,     a1[j]);
    atomicAdd(base + 2 * FDIM + j, a2[j]);
  }
}

// ---------------- pass 2 + trace: h[r,f] += sum_k vals[k,e]*h1[c,k,f] ----------------
__global__ void __launch_bounds__(256) iso_pass2(
    const float* __restrict__ h1, const float* __restrict__ vals,
    const int* __restrict__ rows, const int* __restrict__ cols,
    float* __restrict__ h, int nE, int nRuns) {
  int t   = blockIdx.x * blockDim.x + threadIdx.x;
  int fc  = (t & 15) * 4;
  int run = t >> 4;
  if (run >= nRuns) return;
  int e0 = run * 16;
  int e1 = e0 + 16; if (e1 > nE) e1 = nE;

  float acc[4] = {0,0,0,0};
  int curRow = rows[e0];
  for (int e = e0; e < e1; ++e) {
    int   r  = rows[e];
    int   c  = cols[e];
    float v0 = vals[e];
    float v1 = vals[(long)nE + e];
    float v2 = vals[2L * nE + e];
    const float* hb = h1 + (long)c * 3 * FDIM + fc;
    float4 b0 = *(const float4*)(hb);
    float4 b1 = *(const float4*)(hb + FDIM);
    float4 b2 = *(const float4*)(hb + 2 * FDIM);
    if (e + 1 < e1)
      __builtin_prefetch(h1 + (long)cols[e + 1] * 3 * FDIM + fc, 0, 0);
    if (r != curRow) {
      float* base = h + (long)curRow * FDIM + fc;
      #pragma unroll
      for (int j = 0; j < 4; ++j) { atomicAdd(base + j, acc[j]); acc[j] = 0.f; }
      curRow = r;
    }
    float x0[4] = {b0.x, b0.y, b0.z, b0.w};
    float x1[4] = {b1.x, b1.y, b1.z, b1.w};
    float x2[4] = {b2.x, b2.y, b2.z, b2.w};
    #pragma unroll
    for (int j = 0; j < 4; ++j)
      acc[j] += v0 * x0[j] + v1 * x1[j] + v2 * x2[j];
  }
  float* base = h + (long)curRow * FDIM + fc;
  #pragma unroll
  for (int j = 0; j < 4; ++j) atomicAdd(base + j, acc[j]);
}

// ---------------- GEMM 1: t = h @ W  (full-precision f32 WMMA 16x16x4) ----------------
// block = 128 threads = 4 waves; wave w -> 16-col tile, block -> 16-row tile.
__global__ void __launch_bounds__(128) iso_gemm_xw(
    const float* __restrict__ h, const float* __restrict__ W,
    float* __restrict__ t, int nRows) {
  int rowTile = blockIdx.x * 16;
  int wave    = threadIdx.x >> 5;
  int lane    = threadIdx.x & 31;
  int colTile = wave * 16;
  int m  = lane & 15;
  int hi = lane >> 4;

  int arow = rowTile + m; if (arow >= nRows) arow = nRows - 1;
  const float* hrow = h + (long)arow * FDIM;

  v8f c = {};
  #pragma unroll
  for (int kk = 0; kk < FDIM; kk += 4) {
    int ko = kk + hi * 2;              // A: lane half selects K pair (ISA 16x4 f32 layout)
    v2f a; a.x = hrow[ko];     a.y = hrow[ko + 1];
    v2f b; b.x = W[(long)ko * FDIM + colTile + m];
           b.y = W[(long)(ko + 1) * FDIM + colTile + m];
    c = __builtin_amdgcn_wmma_f32_16x16x4_f32(false, a, false, b, (short)0, c,
                                              false, false);
  }
  #pragma unroll
  for (int r = 0; r < 8; ++r) {
    int row = rowTile + r + 8 * hi;    // C/D layout: VGPR r -> M = r + 8*(lane/16)
    if (row < nRows) t[(long)row * FDIM + colTile + m] = c[r];
  }
}

// ---------------- GEMM 2 (in place on d_out): out = t * tanh(t@Wc + bc) ----------------
__global__ void __launch_bounds__(128) iso_gemm_coeff(
    const float* __restrict__ t_in, const float* __restrict__ Wc,
    const float* __restrict__ bc, float* __restrict__ out, int nRows) {
  int rowTile = blockIdx.x * 16;
  int wave    = threadIdx.x >> 5;
  int lane    = threadIdx.x & 31;
  int colTile = wave * 16;
  int m  = lane & 15;
  int hi = lane >> 4;
  int col = colTile + m;

  int arow = rowTile + m; if (arow >= nRows) arow = nRows - 1;
  const float* trow = t_in + (long)arow * FDIM;

  v8f c = {};
  #pragma unroll
  for (int kk = 0; kk < FDIM; kk += 4) {
    int ko = kk + hi * 2;
    v2f a; a.x = trow[ko];     a.y = trow[ko + 1];
    v2f b; b.x = Wc[(long)ko * FDIM + col];
           b.y = Wc[(long)(ko + 1) * FDIM + col];
    c = __builtin_amdgcn_wmma_f32_16x16x4_f32(false, a, false, b, (short)0, c,
                                              false, false);
  }
  float bias = bc[col];
  float tv[8];
  #pragma unroll
  for (int r = 0; r < 8; ++r) {
    int row = rowTile + r + 8 * hi;
    tv[r] = (row < nRows) ? t_in[(long)row * FDIM + col] : 0.f;
  }
  __syncthreads();   // all reads of t done before any in-place store
  #pragma unroll
  for (int r = 0; r < 8; ++r) {
    int row = rowTile + r + 8 * hi;
    if (row < nRows)
      out[(long)row * FDIM + col] = tv[r] * tanhf(c[r] + bias);
  }
}

extern "C" void kernel_launch(void* const* d_in, const int* in_sizes, int n_in,
                              void* d_out, int out_size, void* d_ws, size_t ws_size,
                              hipStream_t stream) {
  const float* x    = (const float*)d_in[0];
  const float* vals = (const float*)d_in[1];
  const float* W    = (const float*)d_in[2];
  const float* Wc   = (const float*)d_in[3];
  const float* bc   = (const float*)d_in[4];
  const int*   rows = (const int*)d_in[5];
  const int*   cols = (const int*)d_in[6];
  float*       out  = (float*)d_out;

  int E = in_sizes[5];
  int N = in_sizes[0] / FDIM;

  // workspace: h1 [N,3,64] then h [N,64]  (4*N*64 floats = N*64 float4)
  float* h1 = (float*)d_ws;
  float* h  = h1 + (long)N * 3 * FDIM;

  long n4 = (long)N * FDIM;                    // float4 count for h1+h
  iso_zero<<<2048, 256, 0, stream>>>((float4*)d_ws, n4);

  int nRuns   = (E + 15) / 16;
  int blocks1 = (nRuns * 16 + 255) / 256;
  iso_pass1<<<blocks1, 256, 0, stream>>>(x, vals, rows, cols, h1, E, nRuns);
  iso_pass2<<<blocks1, 256, 0, stream>>>(h1, vals, rows, cols, h, E, nRuns);

  int rowTiles = (N + 15) / 16;
  iso_gemm_xw<<<rowTiles, 128, 0, stream>>>(h, W, out, N);
  iso_gemm_coeff<<<rowTiles, 128, 0, stream>>>(out, Wc, bc, out, N);
}